// BinaryTreeCRF_15255723835572
// MI455X (gfx1250) — compile-verified
//
#include <hip/hip_runtime.h>
#include <hip/hip_bf16.h>
#include <math.h>

typedef __attribute__((ext_vector_type(2))) float v2f;
typedef __attribute__((ext_vector_type(8))) float v8f;

#define KDIM 1024
#define LBL  32
#define DEPTH 13
#define MNODES ((1 << (DEPTH + 1)) - 1)   /* 16383 */
#define LEAF_OFF ((1 << DEPTH) - 1)       /* 8191  */

// ---------------------------------------------------------------------------
// Kernel 1: E = hidden @ W^T + b    (16383 x 1024) x (1024 x 32) fp32 WMMA
// One wave per 16-row tile; computes both 16-col output tiles so `hidden`
// is streamed exactly once (memory bound: 67MB @ 23.3TB/s).
// A frag: lane(l): row = l&15, k = k0 + 2*(l>>4) + v   (ISA 32-bit A 16x4)
// B frag: same K-split; col = l&15.  C frag: row = v + 8*(l>>4), col = l&15.
// ---------------------------------------------------------------------------
__global__ void crf_emit_gemm(const float* __restrict__ hidden,
                              const float* __restrict__ W,
                              const float* __restrict__ bias,
                              float* __restrict__ E) {
  const int lane = threadIdx.x;
  const int n16  = lane & 15;
  const int hi   = lane >> 4;
  const int rt   = blockIdx.x;          // 0..1023 row tiles

  int row  = rt * 16 + n16;
  int srow = row < MNODES ? row : (MNODES - 1);   // clamp pad row (loads only)
  const float* Arow = hidden + (size_t)srow * KDIM;
  const float* Bc0  = W + (size_t)n16 * KDIM;        // W^T col n16      = W row n16
  const float* Bc1  = W + (size_t)(16 + n16) * KDIM; // W^T col 16+n16

  v8f c0 = {}; v8f c1 = {};
  for (int k0 = 0; k0 < KDIM; k0 += 4) {
    const int kk = k0 + 2 * hi;
    v2f a  = *(const v2f*)(Arow + kk);
    v2f b0 = *(const v2f*)(Bc0 + kk);
    v2f b1 = *(const v2f*)(Bc1 + kk);
    c0 = __builtin_amdgcn_wmma_f32_16x16x4_f32(false, a, false, b0, (short)0, c0, false, false);
    c1 = __builtin_amdgcn_wmma_f32_16x16x4_f32(false, a, false, b1, (short)0, c1, false, false);
  }
  const float bv0 = bias[n16];
  const float bv1 = bias[16 + n16];
#pragma unroll
  for (int v = 0; v < 8; ++v) {
    const int orow = rt * 16 + v + 8 * hi;
    if (orow < MNODES) {
      E[(size_t)orow * LBL + n16]      = c0[v] + bv0;
      E[(size_t)orow * LBL + 16 + n16] = c1[v] + bv1;
    }
  }
}

// ---------------------------------------------------------------------------
// Kernel 2: ETA = exp(trans), stored row-major (32 x 1024): row i, col j*32+k.
// This 128KB table stays L2-resident for the whole tree contraction.
// ---------------------------------------------------------------------------
__global__ void crf_exp_trans(const float* __restrict__ trans,
                              float* __restrict__ ETA) {
  int idx = blockIdx.x * blockDim.x + threadIdx.x;
  if (idx < LBL * LBL * LBL) ETA[idx] = expf(trans[idx]);
}

// ---------------------------------------------------------------------------
// Kernel 3: one tree level.
//   s[i,n] = sum_{j,k} ET[i,(j,k)] * expL[j,n]*expR[k,n]
//   out[n,i] = Elev[n,i] + maxL[n] + maxR[n] + log(s[i,n])
// One wave per 16-node tile; B fragment built on the fly from LDS-staged
// expL/expR (rank-1 outer product), 512 v_wmma_f32_16x16x4_f32 per wave.
// ---------------------------------------------------------------------------
__global__ void crf_level(const float* __restrict__ scoreIn,  // (2*count) x 32
                          const float* __restrict__ Elev,     // count x 32
                          const float* __restrict__ ETA,      // 32 x 1024
                          float* __restrict__ scoreOut,       // count x 32
                          int count) {
  // odd strides -> conflict-free for lane pattern (n = lane&15, halves +-2 col)
  __shared__ __align__(16) float expLT[16 * 33];   // [n][j]
  __shared__ __align__(16) float expRT[16 * 34];   // [n][k]  (float2 loads)

  const int lane = threadIdx.x;
  const int n = lane & 15;
  const int h = lane >> 4;

  int node = blockIdx.x * 16 + n;
  int cn   = node < count ? node : (count - 1);    // clamp pad lanes (loads)
  const float* Lrow = scoreIn + (size_t)(2 * cn) * LBL;
  const float* Rrow = Lrow + LBL;

  float lv[16], rv[16];
  float mL = -INFINITY, mR = -INFINITY;
#pragma unroll
  for (int t = 0; t < 16; ++t) {
    lv[t] = Lrow[16 * h + t];
    rv[t] = Rrow[16 * h + t];
    mL = fmaxf(mL, lv[t]);
    mR = fmaxf(mR, rv[t]);
  }
  // combine halves: every lane gets the full-vector max for its node
  mL = fmaxf(mL, __shfl_xor(mL, 16, 32));
  mR = fmaxf(mR, __shfl_xor(mR, 16, 32));
#pragma unroll
  for (int t = 0; t < 16; ++t) {
    expLT[n * 33 + 16 * h + t] = expf(lv[t] - mL);
    expRT[n * 34 + 16 * h + t] = expf(rv[t] - mR);
  }
  __syncthreads();

  const float* A0 = ETA + (size_t)n * KDIM;          // rows i = 0..15
  const float* A1 = ETA + (size_t)(16 + n) * KDIM;   // rows i = 16..31

  v8f c0 = {}; v8f c1 = {};
  for (int k0 = 0; k0 < KDIM; k0 += 4) {
    const int kk = k0 + 2 * h;        // even -> (kk,kk+1) share j
    const int j  = kk >> 5;
    const int kx = kk & 31;
    v2f a0 = *(const v2f*)(A0 + kk);
    v2f a1 = *(const v2f*)(A1 + kk);
    const float eL = expLT[n * 33 + j];
    v2f eR = *(const v2f*)(&expRT[n * 34 + kx]);
    v2f b;
    b.x = eL * eR.x;
    b.y = eL * eR.y;
    c0 = __builtin_amdgcn_wmma_f32_16x16x4_f32(false, a0, false, b, (short)0, c0, false, false);
    c1 = __builtin_amdgcn_wmma_f32_16x16x4_f32(false, a1, false, b, (short)0, c1, false, false);
  }

  if (node < count) {
    const float base = mL + mR;
#pragma unroll
    for (int v = 0; v < 8; ++v) {
      const int i0 = v + 8 * h;               // C-frag row
      scoreOut[(size_t)node * LBL + i0] =
          Elev[(size_t)node * LBL + i0] + base + logf(c0[v]);
      scoreOut[(size_t)node * LBL + 16 + i0] =
          Elev[(size_t)node * LBL + 16 + i0] + base + logf(c1[v]);
    }
  }
}

// ---------------------------------------------------------------------------
extern "C" void kernel_launch(void* const* d_in, const int* in_sizes, int n_in,
                              void* d_out, int out_size, void* d_ws, size_t ws_size,
                              hipStream_t stream) {
  const float* hidden = (const float*)d_in[0];   // (16383, 1024)
  const float* W      = (const float*)d_in[1];   // (32, 1024)
  const float* bias   = (const float*)d_in[2];   // (32,)
  const float* trans  = (const float*)d_in[3];   // (32, 32, 32)
  float* out = (float*)d_out;                    // (32,)

  float* E    = (float*)d_ws;                    // 16384*32 floats (2 MB)
  float* ETA  = E + (size_t)16384 * LBL;         // 32768 floats (128 KB)
  float* bufA = ETA + 32768;                     // 4096*32 floats
  float* bufB = bufA + (size_t)4096 * LBL;       // 2048*32 floats

  // 1) emissions
  crf_emit_gemm<<<1024, 32, 0, stream>>>(hidden, W, bias, E);
  // 2) exp(trans) table
  crf_exp_trans<<<(LBL * LBL * LBL + 255) / 256, 256, 0, stream>>>(trans, ETA);

  // 3) tree contraction, leaves = E rows [8191 .. 16382]
  const float* prev = E + (size_t)LEAF_OFF * LBL;
  float* pong[2] = { bufA, bufB };
  int pp = 0;
  for (int d = DEPTH - 1; d >= 0; --d) {
    const int count = 1 << d;
    float* outp = (d == 0) ? out : pong[pp];
    crf_level<<<(count + 15) / 16, 32, 0, stream>>>(
        prev, E + (size_t)((1 << d) - 1) * LBL, ETA, outp, count);
    prev = outp;
    pp ^= 1;
  }
}